// InputableLSTMDecoder_41721312314023
// MI455X (gfx1250) — compile-verified
//
#include <hip/hip_runtime.h>
#include <hip/hip_bf16.h>
#include <math.h>

// Problem constants (from reference)
#define V_  32000
#define E_  512
#define H_  512
#define FIN 512
#define ENC 512
#define B_  32
#define S_  128
#define T_  128

typedef float v2f __attribute__((ext_vector_type(2)));
typedef float v8f __attribute__((ext_vector_type(8)));
typedef unsigned int v4u __attribute__((ext_vector_type(4)));
typedef int v4i __attribute__((ext_vector_type(4)));
typedef int v8i __attribute__((ext_vector_type(8)));

#if defined(__HIP_DEVICE_COMPILE__)
#if !__has_builtin(__builtin_amdgcn_wmma_f32_16x16x4_f32)
#error "wmma_f32_16x16x4_f32 builtin not available on this toolchain"
#endif
#if !__has_builtin(__builtin_amdgcn_tensor_load_to_lds)
#error "tensor_load_to_lds builtin not available on this toolchain"
#endif
#endif

__device__ __forceinline__ v8f wmma_f32(v2f a, v2f b, v8f c) {
  // args: (neg_a, A, neg_b, B, c_mod, C, reuse_a, reuse_b)
  return __builtin_amdgcn_wmma_f32_16x16x4_f32(false, a, false, b, (short)0, c,
                                               false, false);
}

__device__ __forceinline__ float sigf(float x) {
  return 1.0f / (1.0f + __expf(-x));
}

// ---------------------------------------------------------------------------
// TDM 2D tile load: global [tile_h rows x tile_w elems] (fp32, row stride
// row_stride elems) -> LDS at byte offset lds_off, with hardware padding so
// each LDS row has pitch (tile_w + pad) DWORDs.
// pad_interval_code: 4 -> pad every 32 DWORDs, 5 -> every 64 DWORDs.
// pad_amount_code:   3 -> insert 4 DWORDs.
// Descriptor per CDNA5 ISA (D# groups 0/1; 2D => g2=g3=0).
// This toolchain exposes the 6-arg builtin form:
//   (uint32x4 g0, int32x8 g1, int32x4 g2, int32x4 g3, int32x8 ext, i32 cpol)
// ---------------------------------------------------------------------------
__device__ __forceinline__ void tdm_load_2d(unsigned lds_off,
                                            const float* gaddr, int tile_w,
                                            int tile_h, int row_stride,
                                            int pad_interval_code,
                                            int pad_amount_code) {
  unsigned long long ga = (unsigned long long)(uintptr_t)gaddr;
  v4u g0;
  g0.x = 1u;  // count=1 (valid), is_restore=0, gather_mode=0
  g0.y = lds_off;
  g0.z = (unsigned)(ga & 0xFFFFFFFFu);
  g0.w = (unsigned)((ga >> 32) & 0x01FFFFFFu) | 0x80000000u;  // type=2 ("image")

  const unsigned tdim0 = (unsigned)row_stride;  // generous dim0 (>= tile_w)
  const unsigned tdim1 = 0x40000000u;           // generous dim1
  v8i g1;
  g1[0] = (int)((2u << 16)                             // data_size = 4B
                | (1u << 20)                           // pad_enable
                | ((unsigned)pad_interval_code << 22)  // pad interval
                | ((unsigned)pad_amount_code << 25));  // pad amount
  g1[1] = (int)((tdim0 & 0xFFFFu) << 16);              // tensor_dim0[15:0]
  g1[2] = (int)((tdim0 >> 16) | ((tdim1 & 0xFFFFu) << 16));
  g1[3] = (int)((tdim1 >> 16) | ((unsigned)tile_w << 16));  // tile_dim0
  g1[4] = tile_h;                                           // tile_dim1
  g1[5] = row_stride;  // tensor_dim0_stride[31:0]
  g1[6] = 0;
  g1[7] = 0;
  v4i z4 = {0, 0, 0, 0};
  v8i z8 = {0, 0, 0, 0, 0, 0, 0, 0};
  __builtin_amdgcn_tensor_load_to_lds(g0, g1, z4, z4, z8, 0);
}

// ---------------------------------------------------------------------------
// Generic fp32 WMMA GEMM:  C[M,N] = act( A[M,K] * op(B) + bias )
//   BT=true : B given as [N,K] row-major (NT gemm, both operands K-major)
//   BT=false: B given as [K,N] row-major (NN gemm)
// Block tile 64x64, K-chunk 32. Tiles are DMA'd into double-buffered LDS by
// the Tensor Data Mover (issued by wave 0, synchronized via TENSORcnt), while
// all 8 waves stream v_wmma_f32_16x16x4_f32 out of the other buffer.
// Wave computes 16x32 (two 16x16 accumulators). grid = (N/64, M/64, batch).
// PERM remaps output row  m = s*B+b  ->  b*S+s  (B=32, S=128).
// ---------------------------------------------------------------------------
template <bool BT, bool BIAS, bool TANH, bool PERM>
__global__ void __launch_bounds__(256)
gemm_wmma_k(const float* __restrict__ A, size_t sAb, int lda,
            const float* __restrict__ Bm, size_t sBb, int ldb,
            float* __restrict__ C, size_t sCb, int ldc,
            const float* __restrict__ bias, int K) {
  __shared__ __align__(16) float As[2][64 * 36];
  __shared__ __align__(16) float Bs[2][64 * 36];  // NN view: [32][68]

  const int tid  = threadIdx.x;
  const int lane = tid & 31;
  const int wid  = tid >> 5;
  const int p    = lane & 15;
  const int hi   = lane >> 4;
  const int hi2  = hi << 1;

  const int m0 = blockIdx.y * 64;
  const int n0 = blockIdx.x * 64;
  const int z  = blockIdx.z;

  const float* Ag = A + (size_t)z * sAb + (size_t)m0 * lda;
  const float* Bg = Bm + (size_t)z * sBb;
  float*       Cg = C + (size_t)z * sCb;

  const unsigned ldsA0 = (unsigned)(uintptr_t)(void*)&As[0][0];
  const unsigned ldsA1 = (unsigned)(uintptr_t)(void*)&As[1][0];
  const unsigned ldsB0 = (unsigned)(uintptr_t)(void*)&Bs[0][0];
  const unsigned ldsB1 = (unsigned)(uintptr_t)(void*)&Bs[1][0];

  const int msub  = (wid & 3) * 16;   // 0,16,32,48
  const int nbase = (wid >> 2) * 32;  // 0,32

  v8f acc0 = {0.f, 0.f, 0.f, 0.f, 0.f, 0.f, 0.f, 0.f};
  v8f acc1 = acc0;

  const int nchunks = K / 32;

  if (wid == 0) {
    tdm_load_2d(ldsA0, Ag, 32, 64, lda, 4, 3);
    if constexpr (BT)
      tdm_load_2d(ldsB0, Bg + (size_t)n0 * ldb, 32, 64, ldb, 4, 3);
    else
      tdm_load_2d(ldsB0, Bg + n0, 64, 32, ldb, 5, 3);
  }

  for (int i = 0; i < nchunks; ++i) {
    const int k0 = i * 32;
    if (wid == 0) {
      if (i + 1 < nchunks) {
        const int k1 = k0 + 32;
        const unsigned la = ((i + 1) & 1) ? ldsA1 : ldsA0;
        const unsigned lb = ((i + 1) & 1) ? ldsB1 : ldsB0;
        tdm_load_2d(la, Ag + k1, 32, 64, lda, 4, 3);
        if constexpr (BT)
          tdm_load_2d(lb, Bg + (size_t)n0 * ldb + k1, 32, 64, ldb, 4, 3);
        else
          tdm_load_2d(lb, Bg + (size_t)k1 * ldb + n0, 64, 32, ldb, 5, 3);
        // in-order TDM: <=2 outstanding means the pair for chunk i landed
        __builtin_amdgcn_s_wait_tensorcnt(2);
      } else {
        __builtin_amdgcn_s_wait_tensorcnt(0);
      }
    }
    __syncthreads();

    const float* Asb = &As[i & 1][0];
    const float* Bsb = &Bs[i & 1][0];

#pragma unroll
    for (int kk = 0; kk < 32; kk += 4) {
      v2f a, b0, b1;
      a.x = Asb[(msub + p) * 36 + kk + hi2];
      a.y = Asb[(msub + p) * 36 + kk + hi2 + 1];
      if constexpr (BT) {
        b0.x = Bsb[(nbase + p) * 36 + kk + hi2];
        b0.y = Bsb[(nbase + p) * 36 + kk + hi2 + 1];
        b1.x = Bsb[(nbase + 16 + p) * 36 + kk + hi2];
        b1.y = Bsb[(nbase + 16 + p) * 36 + kk + hi2 + 1];
      } else {
        b0.x = Bsb[(kk + hi2) * 68 + nbase + p];
        b0.y = Bsb[(kk + hi2 + 1) * 68 + nbase + p];
        b1.x = Bsb[(kk + hi2) * 68 + nbase + 16 + p];
        b1.y = Bsb[(kk + hi2 + 1) * 68 + nbase + 16 + p];
      }
      acc0 = wmma_f32(a, b0, acc0);
      acc1 = wmma_f32(a, b1, acc1);
    }
    __syncthreads();  // done reading buf (i&1) before TDM overwrites it
  }

#pragma unroll
  for (int v = 0; v < 8; ++v) {
    int mrow = m0 + msub + v + 8 * hi;
    int c0   = n0 + nbase + p;
    int c1   = c0 + 16;
    float x0 = acc0[v];
    float x1 = acc1[v];
    if constexpr (BIAS) { x0 += bias[c0]; x1 += bias[c1]; }
    if constexpr (TANH) { x0 = tanhf(x0); x1 = tanhf(x1); }
    int orow = mrow;
    if constexpr (PERM) orow = (mrow & 31) * S_ + (mrow >> 5);
    Cg[(size_t)orow * ldc + c0] = x0;
    Cg[(size_t)orow * ldc + c1] = x1;
  }
}

// ---------------------------------------------------------------------------
// feed[s*B+b, 0:512]    = embed[tok[b,s], :]
// feed[s*B+b, 512:1024] = inputs[b,s,:]
// ---------------------------------------------------------------------------
__global__ void k_feed(const int* __restrict__ tok,
                       const float* __restrict__ inputs,
                       const float* __restrict__ embed,
                       float* __restrict__ feed) {
  int sb = blockIdx.x;  // s*32 + b
  int s = sb >> 5, b = sb & 31;
  int c4 = threadIdx.x << 2;
  float4 v;
  if (c4 < E_) {
    int t = tok[b * S_ + s];
    v = *(const float4*)(embed + (size_t)t * E_ + c4);
  } else {
    v = *(const float4*)(inputs + (size_t)(b * S_ + s) * FIN + (c4 - E_));
  }
  *(float4*)(feed + (size_t)sb * (E_ + FIN) + c4) = v;
}

__global__ void k_bsum(const float* __restrict__ a, const float* __restrict__ b,
                       float* __restrict__ o) {
  int i = blockIdx.x * 256 + threadIdx.x;
  o[i] = a[i] + b[i];
}

__global__ void k_copyh(const float* __restrict__ Hs, float* __restrict__ cat) {
  int sb = blockIdx.x;
  int c2 = threadIdx.x << 1;
  *(float2*)(cat + (size_t)sb * 1024 + 512 + c2) =
      *(const float2*)(Hs + (size_t)sb * 512 + c2);
}

// ---------------------------------------------------------------------------
// Sequential LSTM scan. 16 persistent workgroups; block `blk` owns hidden
// units [32*blk, 32*blk+32) i.e. gate columns g*512 + 32*blk + jj, g=0..3.
// Per step:  gates_slice[32,128] = h_prev @ w_hh_slice^T + Xih_slice  (WMMA),
// then the elementwise cell update, then a grid-wide monotonic-counter
// barrier (counter zeroed by hipMemsetAsync each launch).
// ---------------------------------------------------------------------------
__global__ void __launch_bounds__(256)
k_scan(const float* __restrict__ Xih, const float* __restrict__ w_hh,
       const float* __restrict__ hzero, float* __restrict__ Hs,
       float* __restrict__ cstate, unsigned int* __restrict__ bar) {
  __shared__ float Gt[32 * 132];
  const int tid  = threadIdx.x;
  const int lane = tid & 31;
  const int wid  = tid >> 5;
  const int p    = lane & 15;
  const int hi   = lane >> 4;
  const int hi2  = hi << 1;
  const int blk  = blockIdx.x;

  const int nsub     = wid * 16;
  const int colLocal = nsub + p;                                       // 0..127
  const int gr = ((colLocal >> 5) << 9) + blk * 32 + (colLocal & 31);  // gate row
  const float* Brow = w_hh + (size_t)gr * H_;

  for (int s = 0; s < S_; ++s) {
    const float* hprev = (s == 0) ? hzero : (Hs + (size_t)(s - 1) * B_ * H_);

    v8f acc0 = {0.f, 0.f, 0.f, 0.f, 0.f, 0.f, 0.f, 0.f};
    v8f acc1 = acc0;
    for (int k0 = 0; k0 < H_; k0 += 4) {
      const float2 av0 = *(const float2*)(hprev + (size_t)p * H_ + k0 + hi2);
      const float2 av1 =
          *(const float2*)(hprev + (size_t)(16 + p) * H_ + k0 + hi2);
      const float2 bv = *(const float2*)(Brow + k0 + hi2);
      v2f a0, a1, b;
      a0.x = av0.x; a0.y = av0.y;
      a1.x = av1.x; a1.y = av1.y;
      b.x = bv.x;   b.y = bv.y;
      acc0 = wmma_f32(a0, b, acc0);  // batch rows 0..15
      acc1 = wmma_f32(a1, b, acc1);  // batch rows 16..31
    }

    const float* Xg = Xih + (size_t)s * B_ * 2048;
#pragma unroll
    for (int v = 0; v < 8; ++v) {
      int b0r = v + 8 * hi;
      int b1r = 16 + v + 8 * hi;
      Gt[b0r * 132 + colLocal] = acc0[v] + Xg[(size_t)b0r * 2048 + gr];
      Gt[b1r * 132 + colLocal] = acc1[v] + Xg[(size_t)b1r * 2048 + gr];
    }
    __syncthreads();

    for (int idx = tid; idx < 1024; idx += 256) {
      int b  = idx >> 5;
      int jj = idx & 31;
      float gi = Gt[b * 132 + jj];
      float gf = Gt[b * 132 + 32 + jj];
      float gg = Gt[b * 132 + 64 + jj];
      float go = Gt[b * 132 + 96 + jj];
      float cprev = (s == 0) ? 0.f : cstate[b * H_ + blk * 32 + jj];
      float c = sigf(gf) * cprev + sigf(gi) * tanhf(gg);
      float h = sigf(go) * tanhf(c);
      cstate[b * H_ + blk * 32 + jj] = c;
      Hs[((size_t)s * B_ + b) * H_ + blk * 32 + jj] = h;
    }
    __syncthreads();

    if (tid == 0) {
      __hip_atomic_fetch_add(bar, 1u, __ATOMIC_RELEASE,
                             __HIP_MEMORY_SCOPE_AGENT);
      unsigned target = 16u * (unsigned)(s + 1);
      while (__hip_atomic_load(bar, __ATOMIC_ACQUIRE,
                               __HIP_MEMORY_SCOPE_AGENT) < target) {
        __builtin_amdgcn_s_sleep(8);
      }
    }
    __syncthreads();
    __threadfence();
  }
}

// ---------------------------------------------------------------------------
// Softmax over T per (b,s) row of scores P[b][s][t] (in place) and
// transposed write of attention weights to out[b][t][s]. One wave per row.
// ---------------------------------------------------------------------------
__global__ void k_softmax_attn(float* __restrict__ P,
                               const unsigned char* __restrict__ mask,
                               float* __restrict__ attnOut) {
  int row  = blockIdx.x * 8 + (threadIdx.x >> 5);  // b*S + s
  int lane = threadIdx.x & 31;
  int b = row >> 7, s = row & 127;

  float vals[4];
  float mx = -3.402823466e38f;
#pragma unroll
  for (int q = 0; q < 4; ++q) {
    int t = lane + (q << 5);
    float v = P[(size_t)row * T_ + t];
    if (mask[t * B_ + b]) v = -3.402823466e38f;
    vals[q] = v;
    mx = fmaxf(mx, v);
  }
  for (int off = 16; off > 0; off >>= 1) mx = fmaxf(mx, __shfl_xor(mx, off, 32));
  float sum = 0.f;
#pragma unroll
  for (int q = 0; q < 4; ++q) {
    vals[q] = __expf(vals[q] - mx);
    sum += vals[q];
  }
  for (int off = 16; off > 0; off >>= 1) sum += __shfl_xor(sum, off, 32);
  float inv = 1.f / sum;
#pragma unroll
  for (int q = 0; q < 4; ++q) {
    int t = lane + (q << 5);
    float pv = vals[q] * inv;
    P[(size_t)row * T_ + t] = pv;
    attnOut[(size_t)b * T_ * S_ + (size_t)t * S_ + s] = pv;
  }
}

// ---------------------------------------------------------------------------
extern "C" void kernel_launch(void* const* d_in, const int* in_sizes, int n_in,
                              void* d_out, int out_size, void* d_ws,
                              size_t ws_size, hipStream_t stream) {
  const int*           tok     = (const int*)d_in[0];
  const float*         inputs  = (const float*)d_in[1];
  const float*         enc     = (const float*)d_in[2];
  const unsigned char* mask    = (const unsigned char*)d_in[3];
  const float*         embed   = (const float*)d_in[4];
  const float*         w_ih    = (const float*)d_in[5];
  const float*         w_hh    = (const float*)d_in[6];
  const float*         b_ih    = (const float*)d_in[7];
  const float*         b_hh    = (const float*)d_in[8];
  const float*         a_in_w  = (const float*)d_in[9];
  const float*         a_out_w = (const float*)d_in[10];
  const float*         fc_w    = (const float*)d_in[11];
  const float*         fc_b    = (const float*)d_in[12];

  float* ws = (float*)d_ws;
  float* hzero = ws;                      ws += B_ * H_;  // zeroed
  unsigned int* bar = (unsigned int*)ws;  ws += 64;       // zeroed
  float* cstate = ws;                     ws += B_ * H_;
  float* bsum   = ws;                     ws += 2048;
  float* feed   = ws;                     ws += (size_t)4096 * 1024;
  float* Xih    = ws;                     ws += (size_t)4096 * 2048;
  float* Hs     = ws;                     ws += (size_t)4096 * 512;
  float* Q      = ws;                     ws += (size_t)4096 * 512;
  float* P      = ws;                     ws += (size_t)B_ * S_ * T_;
  float* cat    = ws;                     ws += (size_t)4096 * 1024;
  float* outb   = ws;                     ws += (size_t)4096 * 512;

  float* logits = (float*)d_out;
  float* attnO  = logits + (size_t)B_ * S_ * V_;

  // zero h0 + barrier counter (graph-replay safe: done every launch)
  (void)hipMemsetAsync(d_ws, 0, (size_t)(B_ * H_ + 64) * sizeof(float), stream);

  k_bsum<<<8, 256, 0, stream>>>(b_ih, b_hh, bsum);
  k_feed<<<4096, 256, 0, stream>>>(tok, inputs, embed, feed);

  // Xih[4096,2048] = feed @ w_ih^T + (b_ih + b_hh)
  gemm_wmma_k<true, true, false, false>
      <<<dim3(2048 / 64, 4096 / 64, 1), 256, 0, stream>>>(
          feed, 0, 1024, w_ih, 0, 1024, Xih, 0, 2048, bsum, 1024);

  // Sequential recurrence (only truly serial work)
  k_scan<<<16, 256, 0, stream>>>(Xih, w_hh, hzero, Hs, cstate, bar);

  // Q[4096,512] = Hs @ attn_in_w^T
  gemm_wmma_k<true, false, false, false>
      <<<dim3(512 / 64, 4096 / 64, 1), 256, 0, stream>>>(
          Hs, 0, 512, a_in_w, 0, 512, Q, 0, 512, nullptr, 512);

  // scores[b][s][t] = Q_b[s,:] . enc_b[t,:]   (batched NT, 32 batches)
  gemm_wmma_k<true, false, false, false>
      <<<dim3(T_ / 64, S_ / 64, B_), 256, 0, stream>>>(
          Q, 512, B_ * 512, enc, 512, B_ * 512, P, (size_t)S_ * T_, T_,
          nullptr, 512);

  k_softmax_attn<<<4096 / 8, 256, 0, stream>>>(P, mask, attnO);

  // ctx -> cat[:, :512]   (batched NN: P_b[S,T] @ enc_b[T,512])
  gemm_wmma_k<false, false, false, false>
      <<<dim3(512 / 64, S_ / 64, B_), 256, 0, stream>>>(
          P, (size_t)S_ * T_, T_, enc, 512, B_ * 512, cat, 1024, B_ * 1024,
          nullptr, T_);

  // cat[:, 512:1024] = h
  k_copyh<<<4096, 256, 0, stream>>>(Hs, cat);

  // out[b*S+s, :] = tanh(cat @ attn_out_w^T)   (row-permuted epilogue)
  gemm_wmma_k<true, false, true, true>
      <<<dim3(512 / 64, 4096 / 64, 1), 256, 0, stream>>>(
          cat, 0, 1024, a_out_w, 0, 1024, outb, 0, 512, nullptr, 1024);

  // logits[4096,32000] = out @ fc_w^T + fc_b   (dominant GEMM: 134 GFLOP)
  gemm_wmma_k<true, true, false, false>
      <<<dim3(V_ / 64, 4096 / 64, 1), 256, 0, stream>>>(
          outb, 0, 512, fc_w, 0, 512, logits, 0, V_, fc_b, 512);
}